// RGDTLayer_5952824672822
// MI455X (gfx1250) — compile-verified
//
#include <hip/hip_runtime.h>

#define NN   100000
#define NE   1600000
#define NR   500
#define DD   128
#define NH   8
#define DFFD 512
#define ALPHA_ 0.15f
#define SLOPE_ 0.2f

typedef __attribute__((ext_vector_type(2))) float v2f;
typedef __attribute__((ext_vector_type(8))) float v8f;

__device__ __forceinline__ v8f wmma4(v2f a, v2f b, v8f c) {
    // D = A(16x4 f32) * B(4x16 f32) + C(16x16 f32)
    return __builtin_amdgcn_wmma_f32_16x16x4_f32(false, a, false, b, (short)0, c, false, false);
}

__device__ __forceinline__ void atomicMaxF(float* addr, float v) {
    if (v >= 0.0f) atomicMax((int*)addr, __float_as_int(v));
    else           atomicMin((unsigned int*)addr, __float_as_uint(v));
}

// ---------------- LayerNorm over ent_feat: one wave per row ----------------
__global__ __launch_bounds__(256) void ln1_kernel(const float* __restrict__ x,
                                                  const float* __restrict__ g,
                                                  const float* __restrict__ b,
                                                  float* __restrict__ out) {
    int tid = threadIdx.x, wave = tid >> 5, lane = tid & 31;
    size_t row = (size_t)blockIdx.x * 8 + wave;
    const float4 v = *(const float4*)(x + row * DD + lane * 4);
    float s  = v.x + v.y + v.z + v.w;
    float sq = v.x*v.x + v.y*v.y + v.z*v.z + v.w*v.w;
    for (int off = 16; off; off >>= 1) {
        s  += __shfl_xor(s,  off, 32);
        sq += __shfl_xor(sq, off, 32);
    }
    float mean = s * (1.0f / DD);
    float var  = sq * (1.0f / DD) - mean * mean;
    float rs   = rsqrtf(var + 1e-5f);
    float4 gv = *(const float4*)(g + lane * 4);
    float4 bv = *(const float4*)(b + lane * 4);
    float4 o;
    o.x = (v.x - mean) * rs * gv.x + bv.x;
    o.y = (v.y - mean) * rs * gv.y + bv.y;
    o.z = (v.z - mean) * rs * gv.z + bv.z;
    o.w = (v.w - mean) * rs * gv.w + bv.w;
    *(float4*)(out + row * DD + lane * 4) = o;
}

// ------- feat = h @ W_ent  (fp32 WMMA, 32-row tile / block, B reused) ------
__global__ __launch_bounds__(256) void gemm_feat_kernel(const float* __restrict__ A,
                                                        const float* __restrict__ W,
                                                        float* __restrict__ C) {
    __shared__ float As[32 * 132];
    int tid = threadIdx.x;
    size_t row0 = (size_t)blockIdx.x * 32;
    // stage A tile (32x128) into LDS, padded stride 132
    {
        const float4* Ag = (const float4*)(A + row0 * DD);
        #pragma unroll
        for (int i = 0; i < 4; ++i) {
            float4 v = Ag[tid + 256 * i];
            int fi = (tid + 256 * i) * 4, r = fi >> 7, c = fi & 127;
            float* p = &As[r * 132 + c];
            p[0] = v.x; p[1] = v.y; p[2] = v.z; p[3] = v.w;
        }
    }
    __syncthreads();
    int wave = tid >> 5, lane = tid & 31;
    int mrow = lane & 15;
    int khalf = (lane >> 4) * 2;
    int mbase = (lane >> 4) * 8;
    int col = wave * 16 + mrow;
    v8f acc0 = {}, acc1 = {};
    for (int kk = 0; kk < DD; kk += 4) {
        int kb = kk + khalf;
        v2f b;  b.x  = W[kb * DD + col];           b.y  = W[(kb + 1) * DD + col];
        v2f a0; a0.x = As[mrow * 132 + kb];        a0.y = As[mrow * 132 + kb + 1];
        v2f a1; a1.x = As[(16 + mrow) * 132 + kb]; a1.y = As[(16 + mrow) * 132 + kb + 1];
        acc0 = wmma4(a0, b, acc0);
        acc1 = wmma4(a1, b, acc1);
    }
    #pragma unroll
    for (int r = 0; r < 8; ++r) {
        C[(row0 + mbase + r) * DD + col]      = acc0[r];
        C[(row0 + 16 + mbase + r) * DD + col] = acc1[r];
    }
}

// ---------------- eh/et per (node, head) -----------------------------------
__global__ __launch_bounds__(256) void node_attn_kernel(const float* __restrict__ feat,
                                                        const float* __restrict__ ah,
                                                        const float* __restrict__ at,
                                                        float* __restrict__ eh,
                                                        float* __restrict__ et) {
    int id = blockIdx.x * 256 + threadIdx.x;      // id < N*8
    int n = id >> 3, hd = id & 7;
    const float4* fp  = (const float4*)(feat + (size_t)n * DD + hd * 16);
    const float4* ahp = (const float4*)(ah + hd * 16);
    const float4* atp = (const float4*)(at + hd * 16);
    float sh = 0.f, st = 0.f;
    #pragma unroll
    for (int i = 0; i < 4; ++i) {
        float4 f = fp[i], a = ahp[i], t = atp[i];
        sh += f.x*a.x + f.y*a.y + f.z*a.z + f.w*a.w;
        st += f.x*t.x + f.y*t.y + f.z*t.z + f.w*t.w;
    }
    eh[id] = sh; et[id] = st;
}

// ---------------- er[r][h] = ((rel @ W_rel) * attn_r).sum(Dh) --------------
__global__ __launch_bounds__(128) void rel_kernel(const float* __restrict__ rel,
                                                  const float* __restrict__ Wr,
                                                  const float* __restrict__ ar,
                                                  float* __restrict__ er) {
    __shared__ float rrow[DD];
    __shared__ float tmp[DD];
    int r = blockIdx.x, j = threadIdx.x;
    rrow[j] = rel[(size_t)r * DD + j];
    __syncthreads();
    float s = 0.f;
    #pragma unroll 8
    for (int k = 0; k < DD; ++k) s += rrow[k] * Wr[k * DD + j];
    tmp[j] = s * ar[j];
    __syncthreads();
    if (j < NH) {
        float t = 0.f;
        #pragma unroll
        for (int i = 0; i < 16; ++i) t += tmp[j * 16 + i];
        er[r * NH + j] = t;
    }
}

// ---------------- init agg=0, m=-inf, ssum=0 -------------------------------
__global__ __launch_bounds__(256) void init_kernel(float* __restrict__ agg,
                                                   float* __restrict__ m,
                                                   float* __restrict__ ssum) {
    int id = blockIdx.x * 256 + threadIdx.x;      // over N*128
    agg[id] = 0.0f;
    if (id < NN * NH) { m[id] = __uint_as_float(0xFF800000u); ssum[id] = 0.0f; }
}

// ---------------- edge scores + leaky relu + segment max -------------------
__global__ __launch_bounds__(256) void edge_score_kernel(const float* __restrict__ eh,
                                                         const float* __restrict__ et,
                                                         const float* __restrict__ er,
                                                         const int* __restrict__ src,
                                                         const int* __restrict__ dst,
                                                         const int* __restrict__ ety,
                                                         float* __restrict__ abuf,
                                                         float* __restrict__ mbuf) {
    int id = blockIdx.x * 256 + threadIdx.x;      // over E*8
    int e = id >> 3, hd = id & 7;
    int s = src[e], d = dst[e], t = ety[e];
    float sc = eh[s * NH + hd] + et[d * NH + hd] + er[t * NH + hd];
    sc = sc > 0.0f ? sc : SLOPE_ * sc;
    abuf[id] = sc;
    atomicMaxF(&mbuf[d * NH + hd], sc);
}

// ---------------- exp + segment sum ---------------------------------------
__global__ __launch_bounds__(256) void edge_exp_kernel(const int* __restrict__ dst,
                                                       const float* __restrict__ mbuf,
                                                       float* __restrict__ abuf,
                                                       float* __restrict__ ssum) {
    int id = blockIdx.x * 256 + threadIdx.x;
    int e = id >> 3, hd = id & 7;
    int d = dst[e];
    float ex = __expf(abuf[id] - mbuf[d * NH + hd]);
    abuf[id] = ex;
    atomicAdd(&ssum[d * NH + hd], ex);
}

// ---------------- normalize ------------------------------------------------
__global__ __launch_bounds__(256) void edge_norm_kernel(const int* __restrict__ dst,
                                                        const float* __restrict__ ssum,
                                                        float* __restrict__ abuf) {
    int id = blockIdx.x * 256 + threadIdx.x;
    int e = id >> 3, hd = id & 7;
    abuf[id] = abuf[id] / ssum[dst[e] * NH + hd];
}

// ---------------- diffusion scatter: 32 lanes per edge ---------------------
__global__ __launch_bounds__(256) void scatter_kernel(const float* __restrict__ fin,
                                                      const float* __restrict__ abuf,
                                                      const int* __restrict__ src,
                                                      const int* __restrict__ dst,
                                                      float* __restrict__ agg) {
    int gid = blockIdx.x * 256 + threadIdx.x;
    int e = gid >> 5, lane = gid & 31;
    int s = src[e], d = dst[e];                    // wave-uniform
    // pull a future gather row from L2 toward the WGP (global_prefetch_b8)
    if (e + 256 < NE) {
        int sp = src[e + 256];
        __builtin_prefetch(fin + (size_t)sp * DD + lane * 4, 0, 3);
    }
    float a = abuf[e * NH + (lane >> 2)];          // head = (lane*4)/16
    const float4 v = *(const float4*)(fin + (size_t)s * DD + lane * 4);
    float* ap = agg + (size_t)d * DD + lane * 4;
    atomicAdd(ap + 0, v.x * a);
    atomicAdd(ap + 1, v.y * a);
    atomicAdd(ap + 2, v.z * a);
    atomicAdd(ap + 3, v.w * a);
}

// ---------------- f = 0.85*agg + 0.15*feat0, and re-zero agg ---------------
__global__ __launch_bounds__(256) void combine_kernel(float* __restrict__ agg,
                                                      const float* __restrict__ feat0,
                                                      float* __restrict__ fout) {
    size_t id = (size_t)blockIdx.x * 256 + threadIdx.x;   // over N*128/4
    float4* ap = ((float4*)agg) + id;
    float4 g = *ap;
    float4 f0 = ((const float4*)feat0)[id];
    float4 r;
    r.x = (1.0f - ALPHA_) * g.x + ALPHA_ * f0.x;
    r.y = (1.0f - ALPHA_) * g.y + ALPHA_ * f0.y;
    r.z = (1.0f - ALPHA_) * g.z + ALPHA_ * f0.z;
    r.w = (1.0f - ALPHA_) * g.w + ALPHA_ * f0.w;
    ((float4*)fout)[id] = r;
    float4 z; z.x = z.y = z.z = z.w = 0.0f;
    *ap = z;
}

// ------ fused tail: rst=f+h; y=LN2(rst); out = relu(y@W1+b1)@W2+b2+rst -----
// 32 rows per block: two 16-row WMMA tiles per wave share the B fragments,
// halving the W1/W2 L2 read traffic per output row.
__global__ __launch_bounds__(256) void ffn_kernel(const float* __restrict__ F,
                                                  const float* __restrict__ Hs,
                                                  const float* __restrict__ g,
                                                  const float* __restrict__ bvec,
                                                  const float* __restrict__ W1,
                                                  const float* __restrict__ b1,
                                                  const float* __restrict__ W2,
                                                  const float* __restrict__ b2,
                                                  float* __restrict__ out) {
    __shared__ float y[32 * 132];
    __shared__ float rst[32 * 132];
    __shared__ float hid[32 * 516];
    int tid = threadIdx.x, wave = tid >> 5, lane = tid & 31;
    size_t row0 = (size_t)blockIdx.x * 32;

    // residual + LN2; each wave handles 4 rows
    #pragma unroll
    for (int rr = 0; rr < 4; ++rr) {
        int row = wave * 4 + rr;
        float4 fv = *(const float4*)(F  + (row0 + row) * DD + lane * 4);
        float4 hv = *(const float4*)(Hs + (row0 + row) * DD + lane * 4);
        float x0 = fv.x + hv.x, x1 = fv.y + hv.y, x2 = fv.z + hv.z, x3 = fv.w + hv.w;
        float s  = x0 + x1 + x2 + x3;
        float sq = x0*x0 + x1*x1 + x2*x2 + x3*x3;
        for (int off = 16; off; off >>= 1) {
            s  += __shfl_xor(s,  off, 32);
            sq += __shfl_xor(sq, off, 32);
        }
        float mean = s * (1.0f / DD);
        float var  = sq * (1.0f / DD) - mean * mean;
        float rs   = rsqrtf(var + 1e-5f);
        float4 gv = *(const float4*)(g + lane * 4);
        float4 bv = *(const float4*)(bvec + lane * 4);
        float* rp = &rst[row * 132 + lane * 4];
        float* yp = &y[row * 132 + lane * 4];
        rp[0] = x0; rp[1] = x1; rp[2] = x2; rp[3] = x3;
        yp[0] = (x0 - mean) * rs * gv.x + bv.x;
        yp[1] = (x1 - mean) * rs * gv.y + bv.y;
        yp[2] = (x2 - mean) * rs * gv.z + bv.z;
        yp[3] = (x3 - mean) * rs * gv.w + bv.w;
    }
    __syncthreads();

    int mrow = lane & 15;
    int khalf = (lane >> 4) * 2;
    int mbase = (lane >> 4) * 8;

    // GEMM1: hidden = relu(y @ W1 + b1); wave w owns cols [64w, 64w+64)
    v8f acc[2][4] = {};
    for (int kk = 0; kk < DD; kk += 4) {
        int kb = kk + khalf;
        v2f a0; a0.x = y[mrow * 132 + kb];        a0.y = y[mrow * 132 + kb + 1];
        v2f a1; a1.x = y[(16 + mrow) * 132 + kb]; a1.y = y[(16 + mrow) * 132 + kb + 1];
        #pragma unroll
        for (int t = 0; t < 4; ++t) {
            int col = wave * 64 + t * 16 + mrow;
            v2f b; b.x = W1[kb * DFFD + col]; b.y = W1[(kb + 1) * DFFD + col];
            acc[0][t] = wmma4(a0, b, acc[0][t]);
            acc[1][t] = wmma4(a1, b, acc[1][t]);
        }
    }
    #pragma unroll
    for (int half = 0; half < 2; ++half) {
        #pragma unroll
        for (int t = 0; t < 4; ++t) {
            int col = wave * 64 + t * 16 + mrow;
            float bb = b1[col];
            #pragma unroll
            for (int r = 0; r < 8; ++r) {
                float v = acc[half][t][r] + bb;
                hid[(half * 16 + mbase + r) * 516 + col] = v > 0.0f ? v : 0.0f;
            }
        }
    }
    __syncthreads();

    // GEMM2: out = hid @ W2 + b2 + rst; wave w owns cols [16w, 16w+16)
    v8f acc2[2] = {};
    int col = wave * 16 + mrow;
    for (int kk = 0; kk < DFFD; kk += 4) {
        int kb = kk + khalf;
        v2f b;  b.x  = W2[kb * DD + col];           b.y  = W2[(kb + 1) * DD + col];
        v2f a0; a0.x = hid[mrow * 516 + kb];        a0.y = hid[mrow * 516 + kb + 1];
        v2f a1; a1.x = hid[(16 + mrow) * 516 + kb]; a1.y = hid[(16 + mrow) * 516 + kb + 1];
        acc2[0] = wmma4(a0, b, acc2[0]);
        acc2[1] = wmma4(a1, b, acc2[1]);
    }
    float bb = b2[col];
    #pragma unroll
    for (int half = 0; half < 2; ++half) {
        #pragma unroll
        for (int r = 0; r < 8; ++r) {
            int m = half * 16 + mbase + r;
            out[(row0 + m) * DD + col] = acc2[half][r] + bb + rst[m * 132 + col];
        }
    }
}

extern "C" void kernel_launch(void* const* d_in, const int* in_sizes, int n_in,
                              void* d_out, int out_size, void* d_ws, size_t ws_size,
                              hipStream_t stream) {
    const float* ent  = (const float*)d_in[0];
    const float* rel  = (const float*)d_in[1];
    const int*   src  = (const int*)d_in[2];
    const int*   dst  = (const int*)d_in[3];
    const int*   ety  = (const int*)d_in[4];
    const float* ln1g = (const float*)d_in[5];
    const float* ln1b = (const float*)d_in[6];
    const float* Went = (const float*)d_in[7];
    const float* Wrel = (const float*)d_in[8];
    const float* ah   = (const float*)d_in[9];
    const float* at   = (const float*)d_in[10];
    const float* ar   = (const float*)d_in[11];
    const float* ln2g = (const float*)d_in[12];
    const float* ln2b = (const float*)d_in[13];
    const float* W1   = (const float*)d_in[14];
    const float* b1   = (const float*)d_in[15];
    const float* W2   = (const float*)d_in[16];
    const float* b2   = (const float*)d_in[17];
    float* out = (float*)d_out;

    // workspace carve-up (floats)
    float* ws    = (float*)d_ws;
    float* h     = ws;                       // N*128
    float* feat  = h    + (size_t)NN * DD;   // N*128
    float* fA    = feat + (size_t)NN * DD;   // N*128
    float* fB    = fA   + (size_t)NN * DD;   // N*128
    float* agg   = fB   + (size_t)NN * DD;   // N*128
    float* abuf  = agg  + (size_t)NN * DD;   // E*8
    float* eh    = abuf + (size_t)NE * NH;   // N*8
    float* et    = eh   + (size_t)NN * NH;   // N*8
    float* mbuf  = et   + (size_t)NN * NH;   // N*8
    float* ssum  = mbuf + (size_t)NN * NH;   // N*8
    float* er    = ssum + (size_t)NN * NH;   // R*8

    ln1_kernel<<<NN / 8, 256, 0, stream>>>(ent, ln1g, ln1b, h);
    gemm_feat_kernel<<<NN / 32, 256, 0, stream>>>(h, Went, feat);
    node_attn_kernel<<<NN * NH / 256, 256, 0, stream>>>(feat, ah, at, eh, et);
    rel_kernel<<<NR, 128, 0, stream>>>(rel, Wrel, ar, er);
    init_kernel<<<NN * DD / 256, 256, 0, stream>>>(agg, mbuf, ssum);
    edge_score_kernel<<<NE * NH / 256, 256, 0, stream>>>(eh, et, er, src, dst, ety, abuf, mbuf);
    edge_exp_kernel<<<NE * NH / 256, 256, 0, stream>>>(dst, mbuf, abuf, ssum);
    edge_norm_kernel<<<NE * NH / 256, 256, 0, stream>>>(dst, ssum, abuf);

    const float* fin = feat;
    float* fout = fA;
    for (int hop = 0; hop < 5; ++hop) {
        scatter_kernel<<<NE * 32 / 256, 256, 0, stream>>>(fin, abuf, src, dst, agg);
        combine_kernel<<<NN * DD / 4 / 256, 256, 0, stream>>>(agg, feat, fout);
        fin = fout;
        fout = (fout == fA) ? fB : fA;
    }

    ffn_kernel<<<NN / 32, 256, 0, stream>>>(fin, h, ln2g, ln2b, W1, b1, W2, b2, out);
}